// GCN_463856468484
// MI455X (gfx1250) — compile-verified
//
#include <hip/hip_runtime.h>
#include <hip/hip_bf16.h>
#include <math.h>

// ---------------------------------------------------------------------------
// GCN (2x GCNConv + linear + log_softmax) for MI455X / gfx1250 (wave32, WMMA)
//
// N=50000 nodes, E=800000 edges, D_in=D_hid=128, C=10.
// Memory plan (float ws):
//   [0]        dinv   : 50000        (also used as deg during accumulation)
//   [50000]    norm   : 800000       (per-edge dinv[src]*dinv[dst])
//   [850000]   hbuf   : 50000*128    (GEMM output h)
//   [7250000]  aggbuf : 50000*128    (scatter accumulator)
// total = 13,650,000 floats = 54.6 MB
// ---------------------------------------------------------------------------

#define N_NODES   50000
#define N_EDGES   800000
#define D_FEAT    128
#define N_CLASSES 10

typedef float v2f __attribute__((ext_vector_type(2)));
typedef float v8f __attribute__((ext_vector_type(8)));

// ---- degree / norm precompute ---------------------------------------------

__global__ void init_deg_kernel(float* __restrict__ deg) {
    int i = blockIdx.x * blockDim.x + threadIdx.x;
    if (i < N_NODES) deg[i] = 1.0f;               // self-loop
}

__global__ void deg_edges_kernel(const int* __restrict__ dst, float* __restrict__ deg) {
    int e = blockIdx.x * blockDim.x + threadIdx.x;
    if (e < N_EDGES) atomicAdd(&deg[dst[e]], 1.0f);
}

__global__ void dinv_kernel(float* __restrict__ deg_dinv) {
    int i = blockIdx.x * blockDim.x + threadIdx.x;
    if (i < N_NODES) deg_dinv[i] = rsqrtf(deg_dinv[i]);   // deg >= 1 always
}

__global__ void edge_norm_kernel(const int* __restrict__ src, const int* __restrict__ dst,
                                 const float* __restrict__ dinv, float* __restrict__ norm) {
    int e = blockIdx.x * blockDim.x + threadIdx.x;
    if (e < N_EDGES) norm[e] = dinv[src[e]] * dinv[dst[e]];
}

// ---- WMMA GEMM: D[M=50000,128] = f(A[50000,128]) @ W[128,128] -------------
// f(a) = relu(a + abias[k]) when abias != nullptr, else identity.
// One wave per 16x16 output tile; block = 8 waves = the 8 N-tiles of one
// 16-row M-tile. 32 x V_WMMA_F32_16X16X4_F32 per tile.

__global__ void __launch_bounds__(256)
gemm128_wmma_kernel(const float* __restrict__ A, const float* __restrict__ W,
                    float* __restrict__ D, const float* __restrict__ abias) {
    const int K = D_FEAT, N = D_FEAT;
    const int lane  = threadIdx.x & 31;
    const int ntile = threadIdx.x >> 5;      // 0..7
    const int mtile = blockIdx.x;            // 0..3124
    const int lm = lane & 15;
    const int hi = lane >> 4;                // 0 or 1

    const float* arow = A + (size_t)(mtile * 16 + lm) * K;
    const float* wcol = W + (size_t)(ntile * 16 + lm);

    v8f acc = {0.f, 0.f, 0.f, 0.f, 0.f, 0.f, 0.f, 0.f};

    if (abias) {
        for (int k0 = 0; k0 < K; k0 += 4) {
            const int ka = k0 + 2 * hi;      // A: M=lm, K=ka..ka+1 ; B: K rows ka..ka+1, N=col
            v2f a, b;
            a.x = fmaxf(arow[ka]     + abias[ka],     0.f);
            a.y = fmaxf(arow[ka + 1] + abias[ka + 1], 0.f);
            b.x = wcol[(size_t)ka * N];
            b.y = wcol[(size_t)(ka + 1) * N];
            acc = __builtin_amdgcn_wmma_f32_16x16x4_f32(false, a, false, b,
                                                        (short)0, acc, false, false);
        }
    } else {
        for (int k0 = 0; k0 < K; k0 += 4) {
            const int ka = k0 + 2 * hi;
            v2f a, b;
            a.x = arow[ka];
            a.y = arow[ka + 1];
            b.x = wcol[(size_t)ka * N];
            b.y = wcol[(size_t)(ka + 1) * N];
            acc = __builtin_amdgcn_wmma_f32_16x16x4_f32(false, a, false, b,
                                                        (short)0, acc, false, false);
        }
    }

    // C/D layout: VGPR v, lanes 0-15 -> M=v, lanes 16-31 -> M=v+8 ; N=lm
    float* drow = D + (size_t)(mtile * 16 + 8 * hi) * N + (size_t)(ntile * 16 + lm);
#pragma unroll
    for (int v = 0; v < 8; ++v)
        drow[(size_t)v * N] = acc[v];
}

// ---- aggregation -----------------------------------------------------------

// agg[i,:] = h[i,:] * dinv[i]^2   (self-loop term; also initializes the buffer)
__global__ void agg_init_kernel(const float* __restrict__ h, const float* __restrict__ dinv,
                                float* __restrict__ agg) {
    int t = blockIdx.x * blockDim.x + threadIdx.x;   // one float4 per thread
    int node = t >> 5, seg = t & 31;
    if (node >= N_NODES) return;
    float w = dinv[node];
    w *= w;
    float4 v = *(const float4*)(h + (size_t)node * D_FEAT + seg * 4);
    v.x *= w; v.y *= w; v.z *= w; v.w *= w;
    *(float4*)(agg + (size_t)node * D_FEAT + seg * 4) = v;
}

// One wave32 per edge: float4 gather of h[src,:], scaled atomic scatter-add
// into agg[dst,:]. 8 edges per 256-thread block.
__global__ void __launch_bounds__(256)
scatter_edges_kernel(const float* __restrict__ h, const int* __restrict__ src,
                     const int* __restrict__ dst, const float* __restrict__ norm,
                     float* __restrict__ agg) {
    int e = blockIdx.x * (blockDim.x >> 5) + (threadIdx.x >> 5);
    if (e >= N_EDGES) return;
    int lane = threadIdx.x & 31;
    int s = src[e], d = dst[e];
    float nrm = norm[e];
    float4 v = *(const float4*)(h + (size_t)s * D_FEAT + lane * 4);
    float* p = agg + (size_t)d * D_FEAT + lane * 4;
    atomicAdd(p + 0, v.x * nrm);
    atomicAdd(p + 1, v.y * nrm);
    atomicAdd(p + 2, v.z * nrm);
    atomicAdd(p + 3, v.w * nrm);
}

// ---- classifier: logits = relu(agg+b2) @ Wl + bl ; log_softmax ------------
// One wave per node row; Wl staged in LDS; shfl_xor all-reduce across lanes.

__global__ void __launch_bounds__(256)
classifier_kernel(const float* __restrict__ agg, const float* __restrict__ b2,
                  const float* __restrict__ Wl, const float* __restrict__ bl,
                  float* __restrict__ out) {
    __shared__ float sW[D_FEAT * N_CLASSES];
    for (int i = threadIdx.x; i < D_FEAT * N_CLASSES; i += blockDim.x) sW[i] = Wl[i];
    __syncthreads();

    int wave = threadIdx.x >> 5;
    int lane = threadIdx.x & 31;
    int row  = blockIdx.x * 8 + wave;          // 50000 = 6250 * 8, no remainder
    if (row >= N_NODES) return;

    int c0 = lane * 4;                          // this lane's 4 feature dims
    float4 v = *(const float4*)(agg + (size_t)row * D_FEAT + c0);
    v.x = fmaxf(v.x + b2[c0 + 0], 0.f);
    v.y = fmaxf(v.y + b2[c0 + 1], 0.f);
    v.z = fmaxf(v.z + b2[c0 + 2], 0.f);
    v.w = fmaxf(v.w + b2[c0 + 3], 0.f);

    float p[N_CLASSES];
#pragma unroll
    for (int c = 0; c < N_CLASSES; ++c) {
        p[c] = v.x * sW[(c0 + 0) * N_CLASSES + c]
             + v.y * sW[(c0 + 1) * N_CLASSES + c]
             + v.z * sW[(c0 + 2) * N_CLASSES + c]
             + v.w * sW[(c0 + 3) * N_CLASSES + c];
    }
#pragma unroll
    for (int m = 16; m >= 1; m >>= 1) {
#pragma unroll
        for (int c = 0; c < N_CLASSES; ++c)
            p[c] += __shfl_xor(p[c], m, 32);
    }
#pragma unroll
    for (int c = 0; c < N_CLASSES; ++c) p[c] += bl[c];

    float mx = p[0];
#pragma unroll
    for (int c = 1; c < N_CLASSES; ++c) mx = fmaxf(mx, p[c]);
    float s = 0.f;
#pragma unroll
    for (int c = 0; c < N_CLASSES; ++c) s += expf(p[c] - mx);
    float lse = mx + logf(s);

    if (lane < N_CLASSES)
        out[(size_t)row * N_CLASSES + lane] = p[lane] - lse;
}

// ---------------------------------------------------------------------------

extern "C" void kernel_launch(void* const* d_in, const int* in_sizes, int n_in,
                              void* d_out, int out_size, void* d_ws, size_t ws_size,
                              hipStream_t stream) {
    const float* x  = (const float*)d_in[0];
    const int*   ei = (const int*)d_in[1];          // [2][E] (int32 under default JAX)
    const float* W1 = (const float*)d_in[2];
    const float* b1 = (const float*)d_in[3];
    const float* W2 = (const float*)d_in[4];
    const float* b2 = (const float*)d_in[5];
    const float* Wl = (const float*)d_in[6];
    const float* bl = (const float*)d_in[7];
    float* out = (float*)d_out;

    const int* src = ei;
    const int* dst = ei + N_EDGES;

    float* ws   = (float*)d_ws;
    float* dinv = ws;                                 // 50000
    float* norm = ws + N_NODES;                       // 800000
    float* hbuf = ws + N_NODES + N_EDGES;             // 6.4M
    float* agg  = hbuf + (size_t)N_NODES * D_FEAT;    // 6.4M

    const int TB = 256;
    const int nodeBlocks  = (N_NODES + TB - 1) / TB;      // 196
    const int edgeBlocks  = N_EDGES / TB;                 // 3125 exact
    const int gemmBlocks  = N_NODES / 16;                 // 3125 exact
    const int aggBlocks   = (N_NODES * 32) / TB;          // 6250 exact
    const int scatBlocks  = N_EDGES / 8;                  // 100000 exact
    const int clsBlocks   = N_NODES / 8;                  // 6250 exact

    // normalization precompute (shared by both conv layers)
    init_deg_kernel<<<nodeBlocks, TB, 0, stream>>>(dinv);
    deg_edges_kernel<<<edgeBlocks, TB, 0, stream>>>(dst, dinv);
    dinv_kernel<<<nodeBlocks, TB, 0, stream>>>(dinv);
    edge_norm_kernel<<<edgeBlocks, TB, 0, stream>>>(src, dst, dinv, norm);

    // layer 1: h = x @ W1 ; agg1 = scatter(h) (bias+relu deferred into GEMM2)
    gemm128_wmma_kernel<<<gemmBlocks, TB, 0, stream>>>(x, W1, hbuf, nullptr);
    agg_init_kernel<<<aggBlocks, TB, 0, stream>>>(hbuf, dinv, agg);
    scatter_edges_kernel<<<scatBlocks, TB, 0, stream>>>(hbuf, src, dst, norm, agg);

    // layer 2: h = relu(agg1 + b1) @ W2 ; agg2 = scatter(h)
    gemm128_wmma_kernel<<<gemmBlocks, TB, 0, stream>>>(agg, W2, hbuf, b1);
    agg_init_kernel<<<aggBlocks, TB, 0, stream>>>(hbuf, dinv, agg);
    scatter_edges_kernel<<<scatBlocks, TB, 0, stream>>>(hbuf, src, dst, norm, agg);

    // classifier + log_softmax: out = log_softmax(relu(agg2 + b2) @ Wl + bl)
    classifier_kernel<<<clsBlocks, TB, 0, stream>>>(agg, b2, Wl, bl, out);
}